// CUDAKernel_52879637348696
// MI455X (gfx1250) — compile-verified
//
#include <hip/hip_runtime.h>

typedef float v2f __attribute__((ext_vector_type(2)));
typedef float v8f __attribute__((ext_vector_type(8)));

#define SEGS 16
#define CHN  128          // U
#define ROW  (SEGS * CHN) // 2048 floats per row
#define NCHUNK 64         // 8 (deg1) + 24 (deg2) + 32 (deg3) chunks of 4 paths

__device__ __forceinline__ float lds_at(const float* base, int byteoff) {
    return *(const float*)((const char*)base + byteoff);
}

// One block per node z. 256 threads = 8 wave32; wave w owns u-tile [16w, 16w+16).
// Scatter-add over paths is a dense f32 WMMA: acc[16s x 16u] += A[16x4] * B[4x16],
// A = one-hot(outseg)*coeff (precomputed per block into atab), B = per-path products.
__global__ __launch_bounds__(256) void tp_wmma_kernel(
    const float* __restrict__ x0,    // [64, 2048]
    const float* __restrict__ x1,    // [Z, 2048]
    const float* __restrict__ c1,    // [32]
    const float* __restrict__ c2,    // [96]
    const float* __restrict__ c3,    // [128]
    const int*   __restrict__ i0,    // [Z]
    const int*   __restrict__ idx1,  // [32, 3]  (x1seg, x0seg, outseg)
    const int*   __restrict__ idx2,  // [96, 4]  (x1seg, x1seg, x0seg, outseg)
    const int*   __restrict__ idx3,  // [128, 5] (x1seg x3, x0seg, outseg)
    float* __restrict__ out)         // [Z, 2048]
{
    __shared__ __align__(16) float  x1s[ROW];
    __shared__ __align__(16) float  x0s[ROW];
    __shared__ __align__(8)  float2 atab[NCHUNK * 32];   // A-matrix per (chunk, lane)
    __shared__ __align__(16) int    r1[32 * 2];          // {j0<<9, i<<9}
    __shared__ __align__(16) int    r2[96 * 4];          // {j0<<9, j1<<9, i<<9, pad}
    __shared__ __align__(16) int    r3[128 * 4];         // {j0<<9, j1<<9, j2<<9, i<<9}

    const int z   = blockIdx.x;
    const int tid = threadIdx.x;

    // ---- stage x1[z] and x0[i0[z]] rows into LDS (8 KB each, float4 moves) ----
    const int row0 = i0[z];
    const float4* __restrict__ gx1 = (const float4*)(x1 + (size_t)z    * ROW);
    const float4* __restrict__ gx0 = (const float4*)(x0 + (size_t)row0 * ROW);
    float4* s1 = (float4*)x1s;
    float4* s0 = (float4*)x0s;
    s1[tid]       = gx1[tid];        // 512 float4 total, 2 per thread
    s1[tid + 256] = gx1[tid + 256];
    s0[tid]       = gx0[tid];
    s0[tid + 256] = gx0[tid + 256];

    // ---- stage packed path records (segment indices pre-scaled to byte offsets) ----
    for (int p = tid; p < 32; p += 256) {
        const int* s = idx1 + p * 3;
        r1[p * 2 + 0] = s[0] << 9;
        r1[p * 2 + 1] = s[1] << 9;
    }
    for (int p = tid; p < 96; p += 256) {
        const int* s = idx2 + p * 4;
        r2[p * 4 + 0] = s[0] << 9;
        r2[p * 4 + 1] = s[1] << 9;
        r2[p * 4 + 2] = s[2] << 9;
        r2[p * 4 + 3] = 0;
    }
    for (int p = tid; p < 128; p += 256) {
        const int* s = idx3 + p * 5;
        r3[p * 4 + 0] = s[0] << 9;
        r3[p * 4 + 1] = s[1] << 9;
        r3[p * 4 + 2] = s[2] << 9;
        r3[p * 4 + 3] = s[3] << 9;
    }

    // ---- precompute A-matrix table: atab[chunk*32 + lane] = (a.x, a.y) ----
    // chunk 0..7   -> deg1 paths 4c..4c+3
    // chunk 8..31  -> deg2 paths 4(c-8)..
    // chunk 32..63 -> deg3 paths 4(c-32)..
    for (int t = tid; t < NCHUNK * 32; t += 256) {
        const int c  = t >> 5;
        const int l  = t & 31;
        const int h  = l >> 4;
        const int nn = l & 15;
        int pA, oa, ob;
        float ca, cb;
        if (c < 8) {
            pA = 4 * c + 2 * h;
            oa = idx1[pA * 3 + 2];       ob = idx1[(pA + 1) * 3 + 2];
            ca = c1[pA];                 cb = c1[pA + 1];
        } else if (c < 32) {
            pA = 4 * (c - 8) + 2 * h;
            oa = idx2[pA * 4 + 3];       ob = idx2[(pA + 1) * 4 + 3];
            ca = c2[pA];                 cb = c2[pA + 1];
        } else {
            pA = 4 * (c - 32) + 2 * h;
            oa = idx3[pA * 5 + 4];       ob = idx3[(pA + 1) * 5 + 4];
            ca = c3[pA];                 cb = c3[pA + 1];
        }
        float2 v;
        v.x = (oa == nn) ? ca : 0.f;
        v.y = (ob == nn) ? cb : 0.f;
        atab[t] = v;
    }
    __syncthreads();

    const int lane = tid & 31;
    const int wave = tid >> 5;                 // 0..7
    const int half = lane >> 4;                // selects K pair {0,1} vs {2,3}
    const int u4   = ((wave << 4) + (lane & 15)) << 2;   // byte offset of this lane's u

    const float2* at = atab + lane;            // stride 32 per chunk

    v8f acc = {0.f, 0.f, 0.f, 0.f, 0.f, 0.f, 0.f, 0.f};

    // ---------------- degree 1 : chunks 0..7 ----------------
#pragma unroll 2
    for (int c = 0; c < 8; ++c) {
        const float2 av = at[c * 32];
        const int4 r = *(const int4*)(r1 + (4 * c + 2 * half) * 2);  // two paths' records
        v2f a = {av.x, av.y};
        v2f b;
        b.x = lds_at(x0s, r.y + u4) * lds_at(x1s, r.x + u4);
        b.y = lds_at(x0s, r.w + u4) * lds_at(x1s, r.z + u4);
        acc = __builtin_amdgcn_wmma_f32_16x16x4_f32(false, a, false, b,
                                                    (short)0, acc, false, false);
    }

    // ---------------- degree 2 : chunks 8..31 ----------------
#pragma unroll 2
    for (int c = 0; c < 24; ++c) {
        const float2 av = at[(8 + c) * 32];
        const int pA = 4 * c + 2 * half;
        const int4 ra = *(const int4*)(r2 + pA * 4);
        const int4 rb = *(const int4*)(r2 + pA * 4 + 4);
        v2f a = {av.x, av.y};
        v2f b;
        b.x = lds_at(x0s, ra.z + u4) * lds_at(x1s, ra.x + u4) * lds_at(x1s, ra.y + u4);
        b.y = lds_at(x0s, rb.z + u4) * lds_at(x1s, rb.x + u4) * lds_at(x1s, rb.y + u4);
        acc = __builtin_amdgcn_wmma_f32_16x16x4_f32(false, a, false, b,
                                                    (short)0, acc, false, false);
    }

    // ---------------- degree 3 : chunks 32..63 ----------------
#pragma unroll 2
    for (int c = 0; c < 32; ++c) {
        const float2 av = at[(32 + c) * 32];
        const int pA = 4 * c + 2 * half;
        const int4 ra = *(const int4*)(r3 + pA * 4);
        const int4 rb = *(const int4*)(r3 + pA * 4 + 4);
        v2f a = {av.x, av.y};
        v2f b;
        b.x = lds_at(x0s, ra.w + u4) * lds_at(x1s, ra.x + u4)
                                     * lds_at(x1s, ra.y + u4) * lds_at(x1s, ra.z + u4);
        b.y = lds_at(x0s, rb.w + u4) * lds_at(x1s, rb.x + u4)
                                     * lds_at(x1s, rb.y + u4) * lds_at(x1s, rb.z + u4);
        acc = __builtin_amdgcn_wmma_f32_16x16x4_f32(false, a, false, b,
                                                    (short)0, acc, false, false);
    }

    // ---- write back: D vgpr i, lane l -> row s = i + 8*half, col u ----
    const int u = (wave << 4) + (lane & 15);
    float* __restrict__ oz = out + (size_t)z * ROW;
#pragma unroll
    for (int i = 0; i < 8; ++i) {
        const int s = i + 8 * half;
        oz[s * CHN + u] = acc[i];
    }
}

extern "C" void kernel_launch(void* const* d_in, const int* in_sizes, int n_in,
                              void* d_out, int out_size, void* d_ws, size_t ws_size,
                              hipStream_t stream) {
    const float* x0   = (const float*)d_in[0];
    const float* x1   = (const float*)d_in[1];
    const float* c1   = (const float*)d_in[2];
    const float* c2   = (const float*)d_in[3];
    const float* c3   = (const float*)d_in[4];
    const int*   i0   = (const int*)  d_in[5];
    const int*   idx1 = (const int*)  d_in[6];
    const int*   idx2 = (const int*)  d_in[7];
    const int*   idx3 = (const int*)  d_in[8];
    float* out = (float*)d_out;

    const int Z = in_sizes[5];   // 4096 nodes, one block each
    tp_wmma_kernel<<<Z, 256, 0, stream>>>(x0, x1, c1, c2, c3, i0, idx1, idx2, idx3, out);
}